// AttnDecoderRNN_5196910428840
// MI455X (gfx1250) — compile-verified
//
#include <hip/hip_runtime.h>
#include <math.h>

#define B_ 128
#define S_ 64
#define T_ 32
#define H_ 512
#define V_ 32000

#define KC 16          // K-chunk staged in LDS per iteration
#define SP 20          // LDS row pitch = KC + 4 (bank-conflict-free for stride-SP b64 reads)

typedef float v2f __attribute__((ext_vector_type(2)));
typedef float v8f __attribute__((ext_vector_type(8)));

// ---------------------------------------------------------------------------
// Generic fp32 WMMA GEMM:  C[m,n] (+)= sum_k A[m,k] * W[n,k]  (+ bias[n])
// A: [M,K] row-major, leading dim lda.  W: [N,K] row-major, leading dim ldw
// (so the WMMA B-matrix is W^T).  C: [M,N], leading dim ldc.
// Block: 256 thr (8 waves) -> 128 N-cols x 64 M-rows.  Each wave: one 16-col
// N tile x four 16-row M tiles.  K staged through LDS in KC-chunks with
// register double-buffering of the next chunk's coalesced float4 loads.
// Requires N%128==0, M%64==0, K%KC==0 (true for every GEMM here).
// ---------------------------------------------------------------------------
__global__ void wmma_gemm_nt(const float* __restrict__ A, long lda,
                             const float* __restrict__ W, long ldw,
                             const float* __restrict__ bias,
                             float* __restrict__ C, long ldc,
                             int K, int accumulate)
{
    __shared__ float As[64 * SP];
    __shared__ float Bs[128 * SP];

    const int tid  = threadIdx.x;
    const int lane = tid & 31;
    const int wave = tid >> 5;
    const int l15  = lane & 15;
    const int half = lane >> 4;     // 0: K-pair {0,1}/rows 0-7 ; 1: K-pair {2,3}/rows 8-15
    const int kb   = half * 2;

    const int n0blk = blockIdx.x * 128;
    const int m0    = blockIdx.y * 64;
    const int nl    = wave * 16;            // wave's N tile inside the block
    const int nCol  = n0blk + nl + l15;

    // staging coordinates: A tile 64x16 (1 float4/thread), B tile 128x16 (2/thread)
    const int sr = tid >> 2;                // 0..63
    const int sc = (tid & 3) * 4;           // 0,4,8,12

    const float* Aptr  = A + (long)(m0 + sr) * lda + sc;
    const float* Bptr0 = W + (long)(n0blk + sr) * ldw + sc;
    const float* Bptr1 = W + (long)(n0blk + sr + 64) * ldw + sc;

    float4 aReg  = *(const float4*)(Aptr);
    float4 bReg0 = *(const float4*)(Bptr0);
    float4 bReg1 = *(const float4*)(Bptr1);

    v8f acc[4];
    if (accumulate) {
        #pragma unroll
        for (int mt = 0; mt < 4; ++mt)
            #pragma unroll
            for (int i = 0; i < 8; ++i)
                acc[mt][i] = C[(long)(m0 + mt * 16 + half * 8 + i) * ldc + nCol];
    } else {
        #pragma unroll
        for (int mt = 0; mt < 4; ++mt)
            #pragma unroll
            for (int i = 0; i < 8; ++i)
                acc[mt][i] = 0.0f;
    }

    for (int kc = 0; kc < K; kc += KC) {
        __syncthreads();                    // previous chunk's compute done
        *(float4*)&As[sr * SP + sc]        = aReg;
        *(float4*)&Bs[sr * SP + sc]        = bReg0;
        *(float4*)&Bs[(sr + 64) * SP + sc] = bReg1;
        if (kc + KC < K) {                  // prefetch next chunk (hidden behind compute)
            aReg  = *(const float4*)(Aptr  + kc + KC);
            bReg0 = *(const float4*)(Bptr0 + kc + KC);
            bReg1 = *(const float4*)(Bptr1 + kc + KC);
        }
        __syncthreads();

        #pragma unroll
        for (int ks = 0; ks < KC; ks += 4) {
            v2f bv = *(const v2f*)&Bs[(nl + l15) * SP + ks + kb];
            #pragma unroll
            for (int mt = 0; mt < 4; ++mt) {
                v2f av = *(const v2f*)&As[(mt * 16 + l15) * SP + ks + kb];
                acc[mt] = __builtin_amdgcn_wmma_f32_16x16x4_f32(
                    false, av, false, bv, (short)0, acc[mt], false, false);
            }
        }
    }

    const float bval = bias ? bias[nCol] : 0.0f;
    #pragma unroll
    for (int mt = 0; mt < 4; ++mt)
        #pragma unroll
        for (int i = 0; i < 8; ++i)
            C[(long)(m0 + mt * 16 + half * 8 + i) * ldc + nCol] = acc[mt][i] + bval;
}

// ---------------------------------------------------------------------------
// Attention for one timestep: scores = Va . tanh(q[b] + Uk[b,s]), softmax,
// ctx = attn @ keys. One block (256 thr, 8 waves) per batch element.
// ---------------------------------------------------------------------------
__global__ void attn_kernel(const float* __restrict__ q,
                            const float* __restrict__ Uk,
                            const float* __restrict__ keys,
                            const float* __restrict__ Va_w,
                            const float* __restrict__ Va_b,
                            float* __restrict__ attn_out,   // [B,T,S]
                            float* __restrict__ ctx,        // [B,H]
                            int t)
{
    __shared__ float qs[H_];
    __shared__ float sc[S_];
    __shared__ float red[S_];

    const int b    = blockIdx.x;
    const int tid  = threadIdx.x;
    const int lane = tid & 31;
    const int wave = tid >> 5;

    for (int j = tid; j < H_; j += 256) qs[j] = q[b * H_ + j];
    __syncthreads();

    // scores: one wave per s (strided)
    for (int s = wave; s < S_; s += 8) {
        const float* u = Uk + (long)(b * S_ + s) * H_;
        float p = 0.0f;
        for (int j = lane; j < H_; j += 32)
            p += Va_w[j] * tanhf(qs[j] + u[j]);
        #pragma unroll
        for (int off = 16; off > 0; off >>= 1)
            p += __shfl_xor(p, off, 32);
        if (lane == 0) sc[s] = p + Va_b[0];
    }
    __syncthreads();

    // softmax over S_=64
    if (tid < S_) red[tid] = sc[tid];
    __syncthreads();
    for (int st = 32; st > 0; st >>= 1) {
        if (tid < st) red[tid] = fmaxf(red[tid], red[tid + st]);
        __syncthreads();
    }
    const float mx = red[0];
    __syncthreads();
    if (tid < S_) { float e = expf(sc[tid] - mx); sc[tid] = e; red[tid] = e; }
    __syncthreads();
    for (int st = 32; st > 0; st >>= 1) {
        if (tid < st) red[tid] += red[tid + st];
        __syncthreads();
    }
    const float inv = 1.0f / red[0];
    __syncthreads();
    if (tid < S_) {
        sc[tid] *= inv;
        attn_out[(long)(b * T_ + t) * S_ + tid] = sc[tid];
    }
    __syncthreads();

    // ctx[b,h] = sum_s attn[s] * keys[b,s,h]
    for (int j = tid; j < H_; j += 256) {
        float a = 0.0f;
        const float* kb = keys + (long)(b * S_) * H_ + j;
        #pragma unroll 4
        for (int s = 0; s < S_; ++s)
            a += sc[s] * kb[(long)s * H_];
        ctx[b * H_ + j] = a;
    }
}

// ---------------------------------------------------------------------------
// GRU elementwise update
// ---------------------------------------------------------------------------
__global__ void gru_kernel(const float* __restrict__ gi,
                           const float* __restrict__ gh,
                           const float* __restrict__ h_old,
                           float* __restrict__ h_new)
{
    const int i = blockIdx.x * blockDim.x + threadIdx.x;
    if (i >= B_ * H_) return;
    const int b = i / H_, j = i % H_;
    const float* gib = gi + (long)b * 3 * H_;
    const float* ghb = gh + (long)b * 3 * H_;
    const float r = 1.0f / (1.0f + expf(-(gib[j]          + ghb[j])));
    const float z = 1.0f / (1.0f + expf(-(gib[H_ + j]     + ghb[H_ + j])));
    const float n = tanhf(gib[2 * H_ + j] + r * ghb[2 * H_ + j]);
    h_new[i] = (1.0f - z) * n + z * h_old[i];
}

// ---------------------------------------------------------------------------
// In-place log_softmax over V for one timestep; one block per batch row.
// ---------------------------------------------------------------------------
__global__ void logsoftmax_kernel(float* __restrict__ lp, int t)
{
    const int b    = blockIdx.x;
    float* row = lp + (long)(b * T_ + t) * V_;
    const int tid  = threadIdx.x;
    const int lane = tid & 31;
    const int wave = tid >> 5;
    __shared__ float red[8];

    float mx = -INFINITY;
    for (int j = tid; j < V_; j += 256) mx = fmaxf(mx, row[j]);
    #pragma unroll
    for (int off = 16; off > 0; off >>= 1) mx = fmaxf(mx, __shfl_xor(mx, off, 32));
    if (lane == 0) red[wave] = mx;
    __syncthreads();
    if (tid == 0) {
        float m = red[0];
        for (int w = 1; w < 8; ++w) m = fmaxf(m, red[w]);
        red[0] = m;
    }
    __syncthreads();
    mx = red[0];

    float sum = 0.0f;
    for (int j = tid; j < V_; j += 256) sum += expf(row[j] - mx);
    #pragma unroll
    for (int off = 16; off > 0; off >>= 1) sum += __shfl_xor(sum, off, 32);
    __syncthreads();
    if (lane == 0) red[wave] = sum;
    __syncthreads();
    if (tid == 0) {
        float s = 0.0f;
        for (int w = 0; w < 8; ++w) s += red[w];
        red[0] = s;
    }
    __syncthreads();
    const float lse = mx + logf(red[0]);
    for (int j = tid; j < V_; j += 256) row[j] -= lse;
}

// ---------------------------------------------------------------------------
// Teacher-forcing embedding gather: embedded[b,t,:] = emb[t==0 ? SOS : tgt[b,t-1]]
// ---------------------------------------------------------------------------
__global__ void embed_kernel(const int* __restrict__ target,
                             const float* __restrict__ emb,
                             float* __restrict__ embedded)
{
    const long i = (long)blockIdx.x * blockDim.x + threadIdx.x;
    if (i >= (long)B_ * T_ * H_) return;
    const int  j  = (int)(i % H_);
    const long bt = i / H_;
    const int  t  = (int)(bt % T_);
    const int  b  = (int)(bt / T_);
    const int tok = (t == 0) ? 0 : target[b * T_ + t - 1];
    embedded[i] = emb[(long)tok * H_ + j];
}

__global__ void copy_kernel(const float* __restrict__ src,
                            float* __restrict__ dst, int n)
{
    const int i = blockIdx.x * blockDim.x + threadIdx.x;
    if (i < n) dst[i] = src[i];
}

// ---------------------------------------------------------------------------
extern "C" void kernel_launch(void* const* d_in, const int* in_sizes, int n_in,
                              void* d_out, int out_size, void* d_ws, size_t ws_size,
                              hipStream_t stream)
{
    const float* keys   = (const float*)d_in[0];   // [B,S,H]
    const float* h0in   = (const float*)d_in[1];   // [1,B,H]
    const int*   target = (const int*)  d_in[2];   // [B,T]
    const float* emb    = (const float*)d_in[3];   // [V,H]
    const float* Wa_w   = (const float*)d_in[4];   // [H,H]
    const float* Wa_b   = (const float*)d_in[5];
    const float* Ua_w   = (const float*)d_in[6];   // [H,H]
    const float* Ua_b   = (const float*)d_in[7];
    const float* Va_w   = (const float*)d_in[8];   // [1,H]
    const float* Va_b   = (const float*)d_in[9];
    const float* W_ih   = (const float*)d_in[10];  // [3H,2H]
    const float* W_hh   = (const float*)d_in[11];  // [3H,H]
    const float* b_ih   = (const float*)d_in[12];
    const float* b_hh   = (const float*)d_in[13];
    const float* out_w  = (const float*)d_in[14];  // [V,H]
    const float* out_b  = (const float*)d_in[15];
    (void)in_sizes; (void)n_in; (void)out_size; (void)ws_size;

    float* out   = (float*)d_out;
    float* lp    = out;                           // log_probs [B,T,V]
    float* hlast = out + (long)B_ * T_ * V_;      // [1,B,H]
    float* attns = hlast + (long)B_ * H_;         // [B,T,S]

    float* ws       = (float*)d_ws;
    float* Uk       = ws; ws += (long)B_ * S_ * H_;   // [B,S,H]
    float* embedded = ws; ws += (long)B_ * T_ * H_;   // [B,T,H]
    float* hbuf0    = ws; ws += B_ * H_;
    float* hbuf1    = ws; ws += B_ * H_;
    float* q        = ws; ws += B_ * H_;
    float* ctx      = ws; ws += B_ * H_;
    float* gi       = ws; ws += B_ * 3 * H_;
    float* gh       = ws; ws += B_ * 3 * H_;

    // setup: h0, embedding gather, Uk = keys @ Ua_w^T + Ua_b
    copy_kernel<<<(B_ * H_ + 255) / 256, 256, 0, stream>>>(h0in, hbuf0, B_ * H_);
    embed_kernel<<<(int)(((long)B_ * T_ * H_ + 255) / 256), 256, 0, stream>>>(
        target, emb, embedded);
    wmma_gemm_nt<<<dim3(H_ / 128, (B_ * S_) / 64), 256, 0, stream>>>(
        keys, H_, Ua_w, H_, Ua_b, Uk, H_, H_, 0);

    const float* hcur = hbuf0;
    float*       hnxt = hbuf1;

    for (int t = 0; t < T_; ++t) {
        // q = h @ Wa_w^T + Wa_b           (M=128, N=512, K=512)
        wmma_gemm_nt<<<dim3(H_ / 128, B_ / 64), 256, 0, stream>>>(
            hcur, H_, Wa_w, H_, Wa_b, q, H_, H_, 0);
        // attention: scores/softmax/ctx
        attn_kernel<<<B_, 256, 0, stream>>>(q, Uk, keys, Va_w, Va_b, attns, ctx, t);
        // gi = e_t @ W_ih[:, :H]^T + b_ih  then  += ctx @ W_ih[:, H:]^T
        wmma_gemm_nt<<<dim3(3 * H_ / 128, B_ / 64), 256, 0, stream>>>(
            embedded + (long)t * H_, (long)T_ * H_, W_ih, 2 * H_, b_ih, gi, 3 * H_, H_, 0);
        wmma_gemm_nt<<<dim3(3 * H_ / 128, B_ / 64), 256, 0, stream>>>(
            ctx, H_, W_ih + H_, 2 * H_, nullptr, gi, 3 * H_, H_, 1);
        // gh = h @ W_hh^T + b_hh
        wmma_gemm_nt<<<dim3(3 * H_ / 128, B_ / 64), 256, 0, stream>>>(
            hcur, H_, W_hh, H_, b_hh, gh, 3 * H_, H_, 0);
        // GRU update
        gru_kernel<<<(B_ * H_ + 255) / 256, 256, 0, stream>>>(gi, gh, hcur, hnxt);
        // logits = h_new @ out_w^T + out_b  directly into log_probs slot
        wmma_gemm_nt<<<dim3(V_ / 128, B_ / 64), 256, 0, stream>>>(
            hnxt, H_, out_w, H_, out_b, lp + (long)t * V_, (long)T_ * V_, H_, 0);
        // in-place log_softmax over V
        logsoftmax_kernel<<<B_, 256, 0, stream>>>(lp, t);

        const float* tmp = hnxt; hnxt = (float*)hcur; hcur = tmp;
    }

    copy_kernel<<<(B_ * H_ + 255) / 256, 256, 0, stream>>>(hcur, hlast, B_ * H_);
}